// STM_38405597561129
// MI455X (gfx1250) — compile-verified
//
#include <hip/hip_runtime.h>

typedef __attribute__((ext_vector_type(2))) float v2f;
typedef __attribute__((ext_vector_type(8))) float v8f;

#define SOM_B 256
#define SOM_D 128
#define SOM_O 4096
// 1 / (SIGMA * sqrt(2*pi)) with SIGMA = 2
#define RADIAL_SCALE 0.19947114020071635f
// -0.5 / SIGMA^2
#define RADIAL_EXPK (-0.125f)

__device__ __forceinline__ unsigned long long shfl_xor_u64(unsigned long long v, int mask) {
    unsigned lo = (unsigned)(v & 0xFFFFFFFFull);
    unsigned hi = (unsigned)(v >> 32);
    lo = __shfl_xor(lo, mask, 32);
    hi = __shfl_xor(hi, mask, 32);
    return ((unsigned long long)hi << 32) | (unsigned long long)lo;
}

// Pass 0: column norms ||w_:,o||^2, row norms ||x_b||^2, init argmin keys.
__global__ __launch_bounds__(256) void som_prep(const float* __restrict__ x,
                                                const float* __restrict__ w,
                                                float* __restrict__ wn,
                                                float* __restrict__ xn,
                                                unsigned long long* __restrict__ keys) {
    const int t = blockIdx.x * blockDim.x + threadIdx.x;
    if (t < SOM_O) {
        float s = 0.0f;
#pragma unroll 8
        for (int d = 0; d < SOM_D; ++d) {
            const float v = w[(size_t)d * SOM_O + t];   // coalesced across t
            s = fmaf(v, v, s);
        }
        wn[t] = s;
    }
    if (t < SOM_B) {
        float s = 0.0f;
#pragma unroll 8
        for (int d = 0; d < SOM_D; ++d) {
            const float v = x[(size_t)t * SOM_D + d];
            s = fmaf(v, v, s);
        }
        xn[t] = s;
        keys[t] = ~0ull;
    }
}

// Pass 1: fp32 WMMA GEMM tile (16x16), fuse norms2 = xn - 2*G + wn, store to out,
// and per-row argmin via monotonic-key atomicMin (value bits << 32 | col index;
// norms2 >= 0 so float bit pattern is order-preserving, low index wins ties).
__global__ __launch_bounds__(256) void som_gemm_argmin(const float* __restrict__ x,
                                                       const float* __restrict__ w,
                                                       const float* __restrict__ xn,
                                                       const float* __restrict__ wn,
                                                       unsigned long long* __restrict__ keys,
                                                       float* __restrict__ out) {
    const int wave = threadIdx.x >> 5;
    const int tile = blockIdx.x * 8 + wave;   // 0..4095 : (16 row tiles) x (256 col tiles)
    const int lane = threadIdx.x & 31;
    const int half = lane >> 4;               // 0: K pair {0,1} / rows 0..7 ; 1: K pair {2,3} / rows 8..15
    const int l15  = lane & 15;

    const int bi = tile >> 8;                 // batch tile  (0..15)
    const int oj = tile & 255;                // output tile (0..255)

    const int arow = bi * 16 + l15;           // A-matrix row M held by this lane
    const int bcol = oj * 16 + l15;           // B/C/D column N held by this lane

    // A frag: lane holds x[arow][k + half*2 + {0,1}]
    const float* ap = x + (size_t)arow * SOM_D + half * 2;
    // B frag: lane holds w[k + half*2 + {0,1}][bcol]
    const float* bp = w + (size_t)(half * 2) * SOM_O + bcol;

    v8f acc = {};
#pragma unroll
    for (int k = 0; k < SOM_D; k += 4) {
        v2f a;
        a.x = ap[k];
        a.y = ap[k + 1];
        v2f b;
        b.x = bp[(size_t)k * SOM_O];
        b.y = bp[(size_t)(k + 1) * SOM_O];
        // D = A(16x4) * B(4x16) + C  -- full fp32 matrix pipe
        acc = __builtin_amdgcn_wmma_f32_16x16x4_f32(
            /*neg_a=*/false, a, /*neg_b=*/false, b,
            /*c_mod=*/(short)0, acc, /*reuse_a=*/false, /*reuse_b=*/false);
    }

    const float wno = wn[bcol];
#pragma unroll
    for (int r = 0; r < 8; ++r) {
        const int gb = bi * 16 + half * 8 + r;              // global batch row of acc[r]
        const float n2 = fmaxf(fmaf(-2.0f, acc[r], xn[gb] + wno), 0.0f);
        out[(size_t)gb * SOM_O + bcol] = n2;

        unsigned long long key =
            ((unsigned long long)__float_as_uint(n2) << 32) | (unsigned)bcol;
        // reduce across the 16 lanes that share this row (xor masks stay in-half)
#pragma unroll
        for (int m = 1; m < 16; m <<= 1) {
            const unsigned long long o = shfl_xor_u64(key, m);
            key = (o < key) ? o : key;
        }
        if (l15 == 0) atomicMin(keys + gb, key);
    }
}

// Pass 2: radial = exp(-0.5 * d2 / sigma^2) / (sigma*sqrt(2pi)); out *= radial.
__global__ __launch_bounds__(256) void som_finalize(float* __restrict__ out,
                                                    const unsigned long long* __restrict__ keys) {
    const int i = blockIdx.x * blockDim.x + threadIdx.x;
    if (i >= SOM_B * SOM_O) return;
    const int b = i >> 12;       // / 4096
    const int o = i & 4095;
    const unsigned wta = (unsigned)(keys[b] & 0xFFFFFFFFull);
    const float wx = (float)(wta >> 6);     // wta // 64
    const float wy = (float)(wta & 63);     // wta %  64
    const float gx = (float)(o >> 6);
    const float gy = (float)(o & 63);
    const float dx = gx - wx;
    const float dy = gy - wy;
    const float d2 = dx * dx + dy * dy;
    const float radial = expf(RADIAL_EXPK * d2) * RADIAL_SCALE;
    out[i] *= radial;
}

extern "C" void kernel_launch(void* const* d_in, const int* in_sizes, int n_in,
                              void* d_out, int out_size, void* d_ws, size_t ws_size,
                              hipStream_t stream) {
    (void)in_sizes; (void)n_in; (void)out_size; (void)ws_size;
    const float* x = (const float*)d_in[0];   // (256, 128)
    const float* w = (const float*)d_in[1];   // (128, 4096)
    float* out = (float*)d_out;               // (256, 4096)

    float* ws_f = (float*)d_ws;
    float* wn = ws_f;                                   // 4096 floats
    float* xn = ws_f + SOM_O;                           // 256 floats
    unsigned long long* keys =
        (unsigned long long*)(ws_f + SOM_O + SOM_B);    // 256 u64, 8B-aligned (offset 17408)

    som_prep<<<16, 256, 0, stream>>>(x, w, wn, xn, keys);
    som_gemm_argmin<<<512, 256, 0, stream>>>(x, w, xn, wn, keys, out);
    som_finalize<<<(SOM_B * SOM_O) / 256, 256, 0, stream>>>(out, keys);
}